// OksuzLSTM_57629871178140
// MI455X (gfx1250) — compile-verified
//
#include <hip/hip_runtime.h>

// ---------------------------------------------------------------------------
// B=32768 scalar LSTMs (H=1) over T=1000, then sigmoid + Linear(1000,2).
// One batch row per lane, 32 rows per wave, 1024 waves total (max parallelism:
// the T-loop is a true recurrence). Latency-bound on the h/c chain, so use the
// hardware v_tanh_f32 trans op when available. The output projection
// ([B,1000]x[1000,2] GEMM) is folded into V_WMMA_F32_16X16X4_F32 accumulation
// every 4 timesteps via xor-16 ds_bpermute lane rearrangement.
// ---------------------------------------------------------------------------

typedef __attribute__((ext_vector_type(2))) float v2f;
typedef __attribute__((ext_vector_type(8))) float v8f;

// Hardware tanh (TRANS32 op on CDNA5) with guarded fallbacks.
#if __has_builtin(__builtin_amdgcn_tanhf)
#define HTANH(x) __builtin_amdgcn_tanhf((x))
#elif __has_builtin(__builtin_amdgcn_tanh_f32)
#define HTANH(x) __builtin_amdgcn_tanh_f32((x))
#else
#define HTANH(x) fmaf(2.0f, __builtin_amdgcn_rcpf(1.0f + __expf(-2.0f * (x))), -1.0f)
#endif

__device__ __forceinline__ float fast_tanh(float x) { return HTANH(x); }

__device__ __forceinline__ float fast_sig(float x) {
#if __has_builtin(__builtin_amdgcn_tanhf) || __has_builtin(__builtin_amdgcn_tanh_f32)
    // sigmoid(x) = 0.5*tanh(0.5x) + 0.5  -> mul + v_tanh + fma
    return fmaf(0.5f, HTANH(0.5f * x), 0.5f);
#else
    // v_exp_f32 + v_rcp_f32
    return __builtin_amdgcn_rcpf(1.0f + __expf(-x));
#endif
}

__device__ __forceinline__ float shfl_x16(float v) {
    int lane = (int)(threadIdx.x & 31u);
    int addr = (lane ^ 16) << 2;
    return __int_as_float(__builtin_amdgcn_ds_bpermute(addr, __float_as_int(v)));
}

#define LSTM_STEP(xv, yOut)                                   \
    do {                                                      \
        float gi = fmaf(h, wh0, fmaf((xv), wi0, bb0));        \
        float gf = fmaf(h, wh1, fmaf((xv), wi1, bb1));        \
        float gg = fmaf(h, wh2, fmaf((xv), wi2, bb2));        \
        float go = fmaf(h, wh3, fmaf((xv), wi3, bb3));        \
        float ii = fast_sig(gi);                              \
        float ff = fast_sig(gf);                              \
        float gt = fast_tanh(gg);                             \
        float oo = fast_sig(go);                              \
        c = fmaf(ff, c, ii * gt);                             \
        h = oo * fast_tanh(c);                                \
        (yOut) = fast_sig(h);                                 \
    } while (0)

__global__ __launch_bounds__(256)
void OksuzLSTM_scan_wmma(const float* __restrict__ x,
                         const float* __restrict__ w_ih,
                         const float* __restrict__ w_hh,
                         const float* __restrict__ b_ih,
                         const float* __restrict__ b_hh,
                         const float* __restrict__ l1_w,
                         const float* __restrict__ l1_b,
                         float* __restrict__ out)
{
    const int lane     = (int)(threadIdx.x & 31u);
    const int row      = blockIdx.x * (int)blockDim.x + (int)threadIdx.x;
    const int waveBase = blockIdx.x * (int)blockDim.x + (int)(threadIdx.x & ~31u);

    // Wave-uniform scalar parameters (lower to s_load).
    const float wi0 = w_ih[0], wi1 = w_ih[1], wi2 = w_ih[2], wi3 = w_ih[3];
    const float wh0 = w_hh[0], wh1 = w_hh[1], wh2 = w_hh[2], wh3 = w_hh[3];
    const float bb0 = b_ih[0] + b_hh[0];
    const float bb1 = b_ih[1] + b_hh[1];
    const float bb2 = b_ih[2] + b_hh[2];
    const float bb3 = b_ih[3] + b_hh[3];

    const float* xrow = x + (size_t)row * 1000u;

    const bool lo   = lane < 16;
    const int  n    = lane & 15;           // WMMA column for this lane
    const int  koff = lo ? 0 : 2;          // K pair held by this lane half
    // Clamped row of l1_w (valid lanes n<2; others load row 0 and are zeroed).
    const float* wrow = l1_w + (size_t)((n < 2) ? n : 0) * 1000u;

    float h = 0.0f, c = 0.0f;
    v8f acc0 = {0.f, 0.f, 0.f, 0.f, 0.f, 0.f, 0.f, 0.f};  // rows waveBase+0..15
    v8f acc1 = {0.f, 0.f, 0.f, 0.f, 0.f, 0.f, 0.f, 0.f};  // rows waveBase+16..31

#pragma unroll 2
    for (int t = 0; t < 1000; t += 4) {
        // Unconditional speculative prefetch ~8 cachelines ahead of the x
        // stream (DEV-scope global_prefetch_b8: OOB translations are dropped).
        __builtin_prefetch(xrow + t + 128, 0, 1);

        const float4 xv = *reinterpret_cast<const float4*>(xrow + t);

        float y0, y1, y2, y3;
        LSTM_STEP(xv.x, y0);
        LSTM_STEP(xv.y, y1);
        LSTM_STEP(xv.z, y2);
        LSTM_STEP(xv.w, y3);

        // Rearrange per-lane {y0..y3} into WMMA 16x4 f32 A-layout:
        //   VGPR0: lanes0-15 K=0, lanes16-31 K=2 ; VGPR1: K=1 / K=3.
        // xor-16 shuffles move the other half-wave's values (no EXEC change).
        const float s0 = shfl_x16(y0);
        const float s1 = shfl_x16(y1);
        const float s2 = shfl_x16(y2);
        const float s3 = shfl_x16(y3);

        v2f a0, a1, bf;
        a0.x = lo ? y0 : s2;   // tile of rows waveBase+0..15
        a0.y = lo ? y1 : s3;
        a1.x = lo ? s0 : y2;   // tile of rows waveBase+16..31
        a1.y = lo ? s1 : y3;

        // B fragment (4x16): B[k][n] = l1_w[n][t+k] for n<2, else 0.
        float bv0 = wrow[t + koff];
        float bv1 = wrow[t + koff + 1];
        bf.x = (n < 2) ? bv0 : 0.0f;
        bf.y = (n < 2) ? bv1 : 0.0f;

        acc0 = __builtin_amdgcn_wmma_f32_16x16x4_f32(
            false, a0, false, bf, (short)0, acc0, false, false);
        acc1 = __builtin_amdgcn_wmma_f32_16x16x4_f32(
            false, a1, false, bf, (short)0, acc1, false, false);
    }

    // D layout: VGPR g holds (M=g, N=lane) for lanes0-15 and (M=g+8, N=lane-16)
    // for lanes16-31. Only columns 0,1 are real outputs. Single base pointer,
    // constant offsets fold into the store immediate.
    if (n < 2) {
        const float bias = l1_b[n];
        float* p = out + ((size_t)(waveBase + (lo ? 0 : 8)) * 2u + (size_t)n);
#pragma unroll
        for (int g = 0; g < 8; ++g) {
            p[2 * g]      = acc0[g] + bias;   // row waveBase + mAdd + g
            p[2 * g + 32] = acc1[g] + bias;   // row waveBase + 16 + mAdd + g
        }
    }
}

extern "C" void kernel_launch(void* const* d_in, const int* in_sizes, int n_in,
                              void* d_out, int out_size, void* d_ws, size_t ws_size,
                              hipStream_t stream) {
    (void)in_sizes; (void)n_in; (void)d_ws; (void)ws_size; (void)out_size;

    const float* x    = (const float*)d_in[0];
    const float* w_ih = (const float*)d_in[1];
    const float* w_hh = (const float*)d_in[2];
    const float* b_ih = (const float*)d_in[3];
    const float* b_hh = (const float*)d_in[4];
    const float* l1_w = (const float*)d_in[5];
    const float* l1_b = (const float*)d_in[6];
    float* out = (float*)d_out;

    const int B = 32768;
    dim3 block(256);
    dim3 grid(B / 256);  // 128 blocks, 1024 waves, one batch row per lane
    hipLaunchKernelGGL(OksuzLSTM_scan_wmma, grid, block, 0, stream,
                       x, w_ih, w_hh, b_ih, b_hh, l1_w, l1_b, out);
}